// VectorQuantizer_56934086475776
// MI455X (gfx1250) — compile-verified
//
#include <hip/hip_runtime.h>

// VectorQuantizer argmax + one-hot for MI455X (gfx1250, wave32).
// z: (16, 512, 64, 64) f32. Output: one_hot(argmax over C) in same layout,
// then 65536 int32 indices appended.
//
// Pure streaming problem: 134MB read + 134MB write -> ~11.5us floor at
// 23.3 TB/s. Strategy: coalesced b128 traffic, async global->LDS double
// buffering (ASYNCcnt pipeline) for the strided channel reads, and
// non-temporal b128 stores for the write-once one-hot output.

typedef float v4f __attribute__((ext_vector_type(4)));
typedef int   v4i __attribute__((ext_vector_type(4)));

#define NCH   512            // channels (argmax axis)
#define HW    4096           // 64*64 spatial per image
#define NB    16             // batch
#define BLOCK 64             // threads per block (2 waves)
#define VEC   4              // positions per thread (one b128)
#define POSB  (BLOCK * VEC)  // 256 positions per block
#define CHUNK 8              // channels staged per async chunk

// ---- CDNA5 async global -> LDS copy (16B per lane), ASYNCcnt tracked ----
// Builtin signature (from hipcc diagnostic): param0 is addrspace(1) v4i*,
// so the call is (global src, lds dst, imm offset, imm cpol).
__device__ __forceinline__ void async_cp16(const float* g, float* l) {
#if __has_builtin(__builtin_amdgcn_global_load_async_to_lds_b128)
  __builtin_amdgcn_global_load_async_to_lds_b128(
      (__attribute__((address_space(1))) v4i*)g,
      (__attribute__((address_space(3))) v4i*)l,
      /*offset=*/0, /*cpol=*/0);
#else
  unsigned lds_addr =
      (unsigned)(size_t)(__attribute__((address_space(3))) float*)l;
  unsigned long long gaddr = (unsigned long long)(size_t)g;
  asm volatile("global_load_async_to_lds_b128 %0, %1, off"
               :: "v"(lds_addr), "v"(gaddr)
               : "memory");
#endif
}

template <int N>
__device__ __forceinline__ void wait_async() {
#if __has_builtin(__builtin_amdgcn_s_wait_asynccnt)
  __builtin_amdgcn_s_wait_asynccnt(N);
#else
  asm volatile("s_wait_asynccnt %0" :: "n"(N) : "memory");
#endif
}

__device__ __forceinline__ void wait_ds0() {
  asm volatile("s_wait_dscnt 0x0" ::: "memory");
}

__global__ __launch_bounds__(BLOCK) void vq_argmax_onehot(
    const float* __restrict__ z, float* __restrict__ zq,
    int* __restrict__ idx_out) {
  // Double-buffered LDS staging: [buf][channel-in-chunk][position-in-block]
  __shared__ float stage[2][CHUNK][POSB];   // 2*8*256*4B = 16 KB

  const int tid = threadIdx.x;
  const int pos = blockIdx.x * POSB + tid * VEC;  // flat (b*HW + hw)
  const int b   = pos >> 12;                      // pos / HW
  const int hw  = pos & (HW - 1);
  const size_t base = (size_t)b * NCH * HW + hw;  // element offset of (b,0,hw)
  const float* src = z + base;                    // channel stride = HW elems

  auto issue = [&](int buf, int c0) {
#pragma unroll
    for (int ch = 0; ch < CHUNK; ++ch)
      async_cp16(src + (size_t)(c0 + ch) * HW, &stage[buf][ch][tid * VEC]);
  };

  v4f m  = {-3.402823466e38f, -3.402823466e38f,
            -3.402823466e38f, -3.402823466e38f};
  v4i mi = {0, 0, 0, 0};

  auto consume = [&](int buf, int c0) {
#pragma unroll
    for (int ch = 0; ch < CHUNK; ++ch) {
      v4f v = *reinterpret_cast<const v4f*>(&stage[buf][ch][tid * VEC]);
      const int c = c0 + ch;
      // strict '>' => first maximum wins, matching jnp.argmax
      if (v.x > m.x) { m.x = v.x; mi.x = c; }
      if (v.y > m.y) { m.y = v.y; mi.y = c; }
      if (v.z > m.z) { m.z = v.z; mi.z = c; }
      if (v.w > m.w) { m.w = v.w; mi.w = c; }
    }
  };

  // Prime the pipeline with two chunks in flight.
  issue(0, 0);
  issue(1, CHUNK);

  // Steady state: wait for oldest chunk (ASYNCcnt <= CHUNK keeps the newer
  // chunk in flight), consume it from LDS, refill its buffer.
  for (int c0 = 0; c0 < NCH - CHUNK; c0 += CHUNK) {
    wait_async<CHUNK>();
    const int buf = (c0 / CHUNK) & 1;
    consume(buf, c0);
    if (c0 + 2 * CHUNK < NCH) {
      wait_ds0();  // all ds_loads of this buffer retired before overwrite
      issue(buf, c0 + 2 * CHUNK);
    }
  }
  // Drain: final chunk needs ASYNCcnt == 0.
  wait_async<0>();
  consume(((NCH - CHUNK) / CHUNK) & 1, NCH - CHUNK);

  // One-hot scatter: write-once stream -> non-temporal b128 stores.
  float* dst = zq + base;
#pragma unroll 4
  for (int c = 0; c < NCH; ++c) {
    v4f o;
    o.x = (c == mi.x) ? 1.0f : 0.0f;
    o.y = (c == mi.y) ? 1.0f : 0.0f;
    o.z = (c == mi.z) ? 1.0f : 0.0f;
    o.w = (c == mi.w) ? 1.0f : 0.0f;
    __builtin_nontemporal_store(o, reinterpret_cast<v4f*>(dst + (size_t)c * HW));
  }

  // Index output (int32), appended after the one-hot tensor.
  __builtin_nontemporal_store(mi, reinterpret_cast<v4i*>(idx_out + pos));
}

extern "C" void kernel_launch(void* const* d_in, const int* in_sizes, int n_in,
                              void* d_out, int out_size, void* d_ws,
                              size_t ws_size, hipStream_t stream) {
  (void)in_sizes; (void)n_in; (void)out_size; (void)d_ws; (void)ws_size;
  const float* z = (const float*)d_in[0];
  float* zq = (float*)d_out;
  int* idx = (int*)(zq + (size_t)NB * NCH * HW);  // tail of d_out
  const int total_pos = NB * HW;                  // 65536
  dim3 grid(total_pos / POSB);                    // 256 blocks
  vq_argmax_onehot<<<grid, BLOCK, 0, stream>>>(z, zq, idx);
}